// TemporalLoss_NoClassWise_Directional_89309549953722
// MI455X (gfx1250) — compile-verified
//
#include <hip/hip_runtime.h>

typedef __attribute__((ext_vector_type(4))) float v4f;
typedef __attribute__((ext_vector_type(2))) float v2f;
typedef __attribute__((ext_vector_type(8))) float v8f;

namespace {
constexpr int   kF      = 4;
constexpr int   kN      = 4;
constexpr int   kC      = 16;
constexpr int   kH      = 512;
constexpr int   kW      = 512;
constexpr int   kPlane  = kH * kW;              // 262144 elements
constexpr int   kFStr   = kN * kC * kPlane;     // elements per frame
constexpr float kEps    = 1e-12f;
constexpr int   kBlock  = 256;                  // 8 waves (wave32)
constexpr int   kQuads  = kN * kH * (kW / 4);   // 262144 float4 positions
constexpr int   kGrid   = kQuads / kBlock;      // 1024 blocks
constexpr float kScale  = 1.0f / (3.0f * kN * kC * kH * kW);  // LOSS_WEIGHT=1
}

__device__ __forceinline__ v4f vrnorm4(v4f ssq) {
  v4f r;
  r.x = 1.0f / fmaxf(sqrtf(ssq.x), kEps);
  r.y = 1.0f / fmaxf(sqrtf(ssq.y), kEps);
  r.z = 1.0f / fmaxf(sqrtf(ssq.z), kEps);
  r.w = 1.0f / fmaxf(sqrtf(ssq.w), kEps);
  return r;
}

__device__ __forceinline__ v4f vabs4(v4f a) {
  v4f r;
  r.x = fabsf(a.x); r.y = fabsf(a.y); r.z = fabsf(a.z); r.w = fabsf(a.w);
  return r;
}

// Stage 1: stream feats once (NT b128 loads), normalize along c in registers,
// accumulate sum of |f_hat[f] - f_hat[f+1]|, reduce to one partial per block.
__global__ __launch_bounds__(kBlock) void tl_partial(
    const float* __restrict__ feats, float* __restrict__ partials) {
  const int tid = blockIdx.x * kBlock + threadIdx.x;   // 0 .. kQuads-1
  const int xq  = tid & (kW / 4 - 1);                  // quad column (128/row)
  const int y   = (tid >> 7) & (kH - 1);
  const int n   = tid >> 16;
  const long base = (long)n * kC * kPlane + (long)y * kW + 4 * xq;

  v4f buf[2][kC];       // ping-pong raw channel tiles (frame f vs f-1)
  v4f rbuf[2];          // 1/max(||.||, eps) per lane-element
  v4f vacc = {0.0f, 0.0f, 0.0f, 0.0f};

#pragma unroll
  for (int f = 0; f < kF; ++f) {
    const v4f* p = (const v4f*)(feats + (long)f * kFStr + base);
    v4f* cur = buf[f & 1];
    v4f ssq = {0.0f, 0.0f, 0.0f, 0.0f};
#pragma unroll
    for (int ch = 0; ch < kC; ++ch) {
      v4f v = __builtin_nontemporal_load(p + ch * (kPlane / 4));  // streaming
      cur[ch] = v;
      ssq += v * v;
    }
    const v4f r = vrnorm4(ssq);
    rbuf[f & 1] = r;
    if (f > 0) {
      const v4f* prev = buf[(f & 1) ^ 1];
      const v4f  rp   = rbuf[(f & 1) ^ 1];
#pragma unroll
      for (int ch = 0; ch < kC; ++ch)
        vacc += vabs4(cur[ch] * r - prev[ch] * rp);
    }
  }
  float s = vacc.x + vacc.y + vacc.z + vacc.w;

  // wave32 tree reduction
#pragma unroll
  for (int off = 16; off > 0; off >>= 1) s += __shfl_down(s, off, 32);

  __shared__ float wsum[kBlock / 32];
  if ((threadIdx.x & 31) == 0) wsum[threadIdx.x >> 5] = s;
  __syncthreads();
  if (threadIdx.x == 0) {
    float b = 0.0f;
#pragma unroll
    for (int i = 0; i < kBlock / 32; ++i) b += wsum[i];
    partials[blockIdx.x] = b;  // deterministic: fixed order, no float atomics
  }
}

// Stage 2: single wave. Each lane serially sums a strided slice of the block
// partials (deterministic), then the 32 lane values are reduced with one
// V_WMMA_F32_16X16X4_F32: A[m][0]=s_m (lanes 0-15), A[m][2]=s_{m+16}
// (lanes 16-31), B = ones  =>  D[m][n] = s_m + s_{m+16}.
__global__ __launch_bounds__(32) void tl_final(
    const float* __restrict__ partials, float* __restrict__ out) {
  const int lane = threadIdx.x;
  float s = 0.0f;
  for (int i = lane; i < kGrid; i += 32) s += partials[i];

  v2f a; a.x = s;    a.y = 0.0f;   // K=0/2 carry data, K=1/3 zero
  v2f b; b.x = 1.0f; b.y = 1.0f;   // all-ones 4x16
  v8f c = {0.0f, 0.0f, 0.0f, 0.0f, 0.0f, 0.0f, 0.0f, 0.0f};
  c = __builtin_amdgcn_wmma_f32_16x16x4_f32(
      /*neg_a=*/false, a, /*neg_b=*/false, b,
      /*c_mod=*/(short)0, c, /*reuse_a=*/false, /*reuse_b=*/false);

  // lane L<16 holds D[0..7][L]; lane L>=16 holds D[8..15][L-16]; rows of D are
  // constant across n, so t_L + t_{L+16} = sum over all 32 lane inputs.
  float t = c[0] + c[1] + c[2] + c[3] + c[4] + c[5] + c[6] + c[7];
  t += __shfl_xor(t, 16, 32);
  if (lane == 0) out[0] = t * kScale;
}

extern "C" void kernel_launch(void* const* d_in, const int* in_sizes, int n_in,
                              void* d_out, int out_size, void* d_ws, size_t ws_size,
                              hipStream_t stream) {
  (void)in_sizes; (void)n_in; (void)out_size; (void)ws_size;
  const float* feats = (const float*)d_in[0];
  // d_in[1] (scores) and d_in[2] (masks) do not affect the forward value:
  // sel-mixing is the identity under stop_gradient; masks are unused.
  float* partials = (float*)d_ws;          // kGrid floats = 4 KB
  float* outp     = (float*)d_out;

  tl_partial<<<kGrid, kBlock, 0, stream>>>(feats, partials);
  tl_final<<<1, 32, 0, stream>>>(partials, outp);
}